// Custom_GCN_21947282883017
// MI455X (gfx1250) — compile-verified
//
#include <hip/hip_runtime.h>

typedef __attribute__((ext_vector_type(2))) float v2f;
typedef __attribute__((ext_vector_type(8))) float v8f;

#define GCN_D 128

// ---------------- degree: deg[dst] += 1 ----------------
__global__ __launch_bounds__(256) void gcn_degree(const int* __restrict__ dst,
                                                  float* __restrict__ deg, int n_edges) {
    int i = blockIdx.x * blockDim.x + threadIdx.x;
    int stride = gridDim.x * blockDim.x;
    for (; i < n_edges; i += stride) atomicAdd(&deg[dst[i]], 1.0f);
}

// ---------------- dinv = rsqrt(1 + deg), in place ----------------
__global__ __launch_bounds__(256) void gcn_dinv(float* __restrict__ deg, int n_nodes) {
    int i = blockIdx.x * blockDim.x + threadIdx.x;
    if (i < n_nodes) deg[i] = rsqrtf(1.0f + deg[i]);
}

// ---------------- H = X * W  (fp32 WMMA 16x16x4) ----------------
// Block = 256 threads = 8 waves; each wave owns a 16-row strip, all 8 col tiles.
// W (128x128 f32 = 64KB) staged in LDS once per block.
__global__ __launch_bounds__(256) void gcn_gemm(const float* __restrict__ X,
                                                const float* __restrict__ W,
                                                float* __restrict__ H, int n) {
    __shared__ float sW[GCN_D * GCN_D];  // 64 KB
    for (int i = threadIdx.x; i < GCN_D * GCN_D; i += blockDim.x) sW[i] = W[i];
    __syncthreads();

    const int lane = threadIdx.x & 31;
    const int wave = threadIdx.x >> 5;
    const int m    = lane & 15;   // row-in-tile for A, col-in-tile for B/D
    const int kh   = lane >> 4;   // K-half selector (0/1)
    const int row0 = blockIdx.x * 128 + wave * 16;

    int rowA = row0 + m;
    if (rowA > n - 1) rowA = n - 1;            // clamp OOB reads; stores guarded below
    const float* xrow = X + (size_t)rowA * GCN_D;

    v8f acc[8] = {};

    for (int kk = 0; kk < 32; ++kk) {          // K = 128 in steps of 4
        const int kb = kk * 4 + kh * 2;
        v2f a;
        a.x = xrow[kb];
        a.y = xrow[kb + 1];
        #pragma unroll
        for (int nt = 0; nt < 8; ++nt) {
            v2f b;
            b.x = sW[(kb + 0) * GCN_D + nt * 16 + m];
            b.y = sW[(kb + 1) * GCN_D + nt * 16 + m];
            acc[nt] = __builtin_amdgcn_wmma_f32_16x16x4_f32(
                false, a, false, b, (short)0, acc[nt], false, false);
        }
    }

    #pragma unroll
    for (int nt = 0; nt < 8; ++nt) {
        #pragma unroll
        for (int v = 0; v < 8; ++v) {
            int r = row0 + v + 8 * kh;         // D layout: M = v + 8*(lane/16)
            if (r < n) H[(size_t)r * GCN_D + nt * 16 + m] = acc[nt][v];
        }
    }
}

// ---------------- edge scatter: agg[dst] += norm * h[src] ----------------
// One wave per edge: 32 lanes x float4 = 128 floats, coalesced 512B gather,
// fp32 global atomics on scatter side.
__global__ __launch_bounds__(256) void gcn_scatter(const int* __restrict__ src,
                                                   const int* __restrict__ dst,
                                                   const float* __restrict__ dinv,
                                                   const float* __restrict__ h,
                                                   float* __restrict__ agg, int n_edges) {
    const int lane = threadIdx.x & 31;
    int wid = blockIdx.x * (blockDim.x >> 5) + (threadIdx.x >> 5);
    const int nw = gridDim.x * (blockDim.x >> 5);
    for (int e = wid; e < n_edges; e += nw) {
        const int s = src[e];
        const int d = dst[e];
        const float norm = dinv[s] * dinv[d];
        const float4 v = ((const float4*)(h + (size_t)s * GCN_D))[lane];
        float* a = agg + (size_t)d * GCN_D + lane * 4;
        atomicAdd(a + 0, v.x * norm);
        atomicAdd(a + 1, v.y * norm);
        atomicAdd(a + 2, v.z * norm);
        atomicAdd(a + 3, v.w * norm);
    }
}

// ---------------- out = agg + h*dinv^2 + b, optional ReLU ----------------
__global__ __launch_bounds__(256) void gcn_combine(const float* __restrict__ h,
                                                   const float* __restrict__ agg,
                                                   const float* __restrict__ dinv,
                                                   const float* __restrict__ bias,
                                                   float* __restrict__ out, int n, int do_relu) {
    size_t i = (size_t)blockIdx.x * blockDim.x + threadIdx.x;  // one float4 each
    size_t total = (size_t)n * (GCN_D / 4);
    if (i >= total) return;
    int node = (int)(i >> 5);   // 32 float4 per node
    int f4   = (int)(i & 31);
    float di = dinv[node];
    float d2 = di * di;
    float4 hv = ((const float4*)h)[i];
    float4 av = ((const float4*)agg)[i];
    float4 bv = ((const float4*)bias)[f4];
    float4 o;
    o.x = av.x + hv.x * d2 + bv.x;
    o.y = av.y + hv.y * d2 + bv.y;
    o.z = av.z + hv.z * d2 + bv.z;
    o.w = av.w + hv.w * d2 + bv.w;
    if (do_relu) {
        o.x = fmaxf(o.x, 0.0f);
        o.y = fmaxf(o.y, 0.0f);
        o.z = fmaxf(o.z, 0.0f);
        o.w = fmaxf(o.w, 0.0f);
    }
    ((float4*)out)[i] = o;
}

extern "C" void kernel_launch(void* const* d_in, const int* in_sizes, int n_in,
                              void* d_out, int out_size, void* d_ws, size_t ws_size,
                              hipStream_t stream) {
    const float* x  = (const float*)d_in[0];
    const int*   ei = (const int*)d_in[1];
    const float* W1 = (const float*)d_in[2];
    const float* b1 = (const float*)d_in[3];
    const float* W2 = (const float*)d_in[4];
    const float* b2 = (const float*)d_in[5];
    const float* W3 = (const float*)d_in[6];
    const float* b3 = (const float*)d_in[7];

    const int n = in_sizes[0] / GCN_D;
    const int E = in_sizes[1] / 2;
    const int* srcI = ei;
    const int* dstI = ei + E;

    // workspace layout: dinv | h | agg | bufA
    float* ws = (float*)d_ws;
    size_t n4 = ((size_t)n + 3) & ~(size_t)3;   // keep 16B alignment downstream
    float* dinv = ws;
    float* h    = ws + n4;
    float* agg  = h + (size_t)n * GCN_D;
    float* bufA = agg + (size_t)n * GCN_D;
    float* out  = (float*)d_out;

    // degrees -> dinv
    hipMemsetAsync(dinv, 0, (size_t)n * sizeof(float), stream);
    gcn_degree<<<1024, 256, 0, stream>>>(dstI, dinv, E);
    gcn_dinv<<<(n + 255) / 256, 256, 0, stream>>>(dinv, n);

    const float* Ws[3] = {W1, W2, W3};
    const float* bs[3] = {b1, b2, b3};
    const float* cur = x;
    for (int l = 0; l < 3; ++l) {
        float* o = (l == 2) ? out : bufA;
        gcn_gemm<<<(n + 127) / 128, 256, 0, stream>>>(cur, Ws[l], h, n);
        hipMemsetAsync(agg, 0, (size_t)n * GCN_D * sizeof(float), stream);
        gcn_scatter<<<2048, 256, 0, stream>>>(srcI, dstI, dinv, h, agg, E);
        size_t total4 = (size_t)n * (GCN_D / 4);
        gcn_combine<<<(int)((total4 + 255) / 256), 256, 0, stream>>>(
            h, agg, dinv, bs[l], o, n, (l < 2) ? 1 : 0);
        cur = o;
    }
}